// SidechainMapCoordinateLoss_62165356642814
// MI455X (gfx1250) — compile-verified
//
#include <hip/hip_runtime.h>
#include <hip/hip_bf16.h>
#include <math.h>

// ---- problem constants (reference shapes are fixed) ----
#define BATCH   64
#define LATOMS  512
#define NRES    1024
#define RATOMS  14
#define CUTOFF_F 5.0f
#define EPSF     1e-12f

typedef __attribute__((ext_vector_type(2))) float v2f;
typedef __attribute__((ext_vector_type(8))) float v8f;

// Workspace layout (floats):
//  [0      .. 2047 ]  ligand-dist num partials   (b*32 + rowtile)
//  [2048   .. 4095 ]  ligand-dist den partials
//  [4096   .. 4159 ]  ligand coord per-batch sum-se
//  [4160   .. 4223 ]  ligand coord per-batch count
//  [4224   .. 5119 ]  sidechain coord mse partials (896 blocks)
//  [5120   .. 6015 ]  sidechain coord cnt partials
//  [6016   .. 14207]  sidechain-dist loss partials (8192 blocks)
//  [14208  .. 22399]  sidechain-dist has  partials
#define WS_LDN   0
#define WS_LDD   2048
#define WS_LSE   4096
#define WS_LCNT  4160
#define WS_SCM   4224
#define WS_SCC   5120
#define WS_SDL   6016
#define WS_SDH   14208

// single v_sqrt_f32 (TRANS pipe, ~1 ULP) instead of the IEEE expansion;
// inputs are clamped to >= 1e-12 so the denorm-rescale path is unnecessary.
static __device__ __forceinline__ float fast_sqrt(float x) {
#if __has_builtin(__builtin_amdgcn_sqrtf)
    return __builtin_amdgcn_sqrtf(x);
#else
    return sqrtf(x);
#endif
}

// One V_WMMA_F32_16X16X4_F32 produces a full 16x16 squared-distance tile:
// A row m = [x, y, z, ||p_m||^2], B col n = [-2x, -2y, -2z, 1], C = ||p_n||^2
static __device__ __forceinline__ v8f wmma_sqdist(v2f a, v2f b, float col_norm) {
    v8f c;
#pragma unroll
    for (int i = 0; i < 8; ++i) c[i] = col_norm;
    return __builtin_amdgcn_wmma_f32_16x16x4_f32(false, a, false, b, (short)0, c,
                                                 false, false);
}

// fixed-order block tree-sum; result broadcast to all threads
static __device__ __forceinline__ float block_sum256(float v, float* red) {
    const int t = threadIdx.x;
    red[t] = v;
    __syncthreads();
#pragma unroll
    for (int off = 128; off > 0; off >>= 1) {
        if (t < off) red[t] += red[t + off];
        __syncthreads();
    }
    float r = red[0];
    __syncthreads();
    return r;
}

// ---------------- ligand coord loss partials (64 blocks x 256) ----------------
__global__ void lig_coord_kernel(const float* __restrict__ pred,
                                 const float* __restrict__ tgt,
                                 const unsigned char* __restrict__ lm,
                                 float* __restrict__ se_out,
                                 float* __restrict__ cnt_out) {
    const int b = blockIdx.x;
    const int t = threadIdx.x;
    float se = 0.f, cnt = 0.f;
    for (int l = t; l < LATOMS; l += 256) {
        const int idx = b * LATOMS + l;
        const float m  = lm[idx] ? 1.f : 0.f;
        const float dx = pred[idx * 3 + 0] - tgt[idx * 3 + 0];
        const float dy = pred[idx * 3 + 1] - tgt[idx * 3 + 1];
        const float dz = pred[idx * 3 + 2] - tgt[idx * 3 + 2];
        se += (dx * dx + dy * dy + dz * dz) * m;
        cnt += m;
    }
    __shared__ float r1[256];
    __shared__ float r2[256];
    const float S = block_sum256(se, r1);
    const float C = block_sum256(cnt, r2);
    if (t == 0) { se_out[b] = S; cnt_out[b] = C; }
}

// ------------- ligand distance loss: WMMA tiles (grid 32x64, 128 thr) ---------
__global__ void lig_dist_kernel(const float* __restrict__ pred,
                                const float* __restrict__ tgt,
                                const unsigned char* __restrict__ lm,
                                float* __restrict__ num_out,
                                float* __restrict__ den_out) {
    const int b    = blockIdx.y;
    const int it   = blockIdx.x;           // row tile 0..31
    const int wave = threadIdx.x >> 5;     // 4 waves
    const int lane = threadIdx.x & 31;
    const int i0   = it * 16;
    const int m    = lane & 15;
    const int hi   = lane >> 4;            // 0: K=0,1  1: K=2,3 half

    // --- A operands (row block), shared by all column tiles ---
    const int ra  = b * LATOMS + i0 + m;
    const float px = pred[ra * 3 + 0], py = pred[ra * 3 + 1], pz = pred[ra * 3 + 2];
    const float tx = tgt [ra * 3 + 0], ty = tgt [ra * 3 + 1], tz = tgt [ra * 3 + 2];
    const float np = px * px + py * py + pz * pz;
    const float nt = tx * tx + ty * ty + tz * tz;
    v2f a_p; a_p[0] = hi ? pz : px; a_p[1] = hi ? np : py;
    v2f a_t; a_t[0] = hi ? tz : tx; a_t[1] = hi ? nt : ty;

    // row masks for the 8 D-matrix rows this lane owns (M = r + 8*hi)
    float rmask[8];
#pragma unroll
    for (int r = 0; r < 8; ++r)
        rmask[r] = lm[b * LATOMS + i0 + r + 8 * hi] ? 1.f : 0.f;

    float num = 0.f, den = 0.f;
    // unroll x2: interleaves two tiles' WMMAs with the other tile's VALU
    // post-processing, filling the WMMA->VALU hazard slots with real work
#pragma unroll 2
    for (int s = 0; s < 8; ++s) {
        const int jt = wave + 4 * s;       // column tile
        const int jc = jt * 16 + m;        // this lane's column atom
        const int ca = b * LATOMS + jc;
        const float qx = pred[ca * 3 + 0], qy = pred[ca * 3 + 1], qz = pred[ca * 3 + 2];
        const float ux = tgt [ca * 3 + 0], uy = tgt [ca * 3 + 1], uz = tgt [ca * 3 + 2];
        const float nq = qx * qx + qy * qy + qz * qz;
        const float nu = ux * ux + uy * uy + uz * uz;
        v2f b_p; b_p[0] = hi ? -2.f * qz : -2.f * qx; b_p[1] = hi ? 1.f : -2.f * qy;
        v2f b_t; b_t[0] = hi ? -2.f * uz : -2.f * ux; b_t[1] = hi ? 1.f : -2.f * uy;

        const v8f sqp = wmma_sqdist(a_p, b_p, nq);
        const v8f sqt = wmma_sqdist(a_t, b_t, nu);

        const float cmask = lm[ca] ? 1.f : 0.f;
#pragma unroll
        for (int r = 0; r < 8; ++r) {
            const int row  = i0 + r + 8 * hi;
            const float tsq = sqt[r];
            const float td  = fast_sqrt(fmaxf(tsq, EPSF));
            const float pd  = fast_sqrt(fmaxf(sqp[r], EPSF));
            const float w   = rmask[r] * cmask;
            const bool ok   = (row != jc) && (tsq > 0.f) && (td <= CUTOFF_F) && (w != 0.f);
            if (ok) {
                const float d = pd - td;
                num += d * d;
                den += 1.f;
            }
        }
    }
    // wave32 reduce, then combine 4 waves
#pragma unroll
    for (int off = 16; off > 0; off >>= 1) {
        num += __shfl_xor(num, off, 32);
        den += __shfl_xor(den, off, 32);
    }
    __shared__ float sn[4];
    __shared__ float sd[4];
    if (lane == 0) { sn[wave] = num; sd[wave] = den; }
    __syncthreads();
    if (threadIdx.x == 0) {
        num_out[b * 32 + it] = sn[0] + sn[1] + sn[2] + sn[3];
        den_out[b * 32 + it] = sd[0] + sd[1] + sd[2] + sd[3];
    }
}

// ----------- sidechain coord loss partials (896 blocks x 256, 4 atoms/thr) ----
__global__ void sc_coord_kernel(const float* __restrict__ pred,
                                const float* __restrict__ tgt,
                                const unsigned char* __restrict__ am,
                                float* __restrict__ mse_out,
                                float* __restrict__ cnt_out) {
    const int gtid = blockIdx.x * 256 + threadIdx.x;
    float s = 0.f, c = 0.f;
#pragma unroll
    for (int k = 0; k < 4; ++k) {
        const int a = gtid + k * (896 * 256);   // 4*896*256 == 64*1024*14
        const float mk = am[a] ? 1.f : 0.f;
        const float dx = pred[a * 3 + 0] - tgt[a * 3 + 0];
        const float dy = pred[a * 3 + 1] - tgt[a * 3 + 1];
        const float dz = pred[a * 3 + 2] - tgt[a * 3 + 2];
        s += (dx * dx + dy * dy + dz * dz) * (1.f / 3.f) * mk;
        c += mk;
    }
    __shared__ float r1[256];
    __shared__ float r2[256];
    const float S = block_sum256(s, r1);
    const float C = block_sum256(c, r2);
    if (threadIdx.x == 0) { mse_out[blockIdx.x] = S; cnt_out[blockIdx.x] = C; }
}

// -------- sidechain distance loss: one wave per residue (8192 blocks x 256) ---
__global__ void sc_dist_kernel(const float* __restrict__ pred,
                               const float* __restrict__ tgt,
                               const unsigned char* __restrict__ am,
                               float* __restrict__ loss_out,
                               float* __restrict__ has_out) {
    const int wave = threadIdx.x >> 5;
    const int lane = threadIdx.x & 31;
    const int res  = blockIdx.x * 8 + wave;     // 0..65535
    const int m    = lane & 15;
    const int hi   = lane >> 4;

    const int base  = res * RATOMS;
    const int aIdx  = (m < RATOMS) ? m : (RATOMS - 1);   // clamp pad rows
    const float cmask = (m < RATOMS && am[base + aIdx]) ? 1.f : 0.f;
    const int cb = (base + aIdx) * 3;

    const float px = pred[cb + 0], py = pred[cb + 1], pz = pred[cb + 2];
    const float tx = tgt [cb + 0], ty = tgt [cb + 1], tz = tgt [cb + 2];
    const float np = px * px + py * py + pz * pz;
    const float nt = tx * tx + ty * ty + tz * tz;

    // residue tile is symmetric: same lane data builds A, B and C
    v2f a_p; a_p[0] = hi ? pz : px; a_p[1] = hi ? np : py;
    v2f a_t; a_t[0] = hi ? tz : tx; a_t[1] = hi ? nt : ty;
    v2f b_p; b_p[0] = hi ? -2.f * pz : -2.f * px; b_p[1] = hi ? 1.f : -2.f * py;
    v2f b_t; b_t[0] = hi ? -2.f * tz : -2.f * tx; b_t[1] = hi ? 1.f : -2.f * ty;

    const v8f sqp = wmma_sqdist(a_p, b_p, np);
    const v8f sqt = wmma_sqdist(a_t, b_t, nt);

    float rmask[8];
#pragma unroll
    for (int r = 0; r < 8; ++r) {
        const int rr = r + 8 * hi;
        rmask[r] = (rr < RATOMS && am[base + ((rr < RATOMS) ? rr : 0)]) ? 1.f : 0.f;
    }

    float num = 0.f, den = 0.f;
#pragma unroll
    for (int r = 0; r < 8; ++r) {
        const int rr = r + 8 * hi;
        const float tsq = sqt[r];
        const float td  = fast_sqrt(fmaxf(tsq, EPSF));
        const float pd  = fast_sqrt(fmaxf(sqp[r], EPSF));
        const float w   = rmask[r] * cmask;
        const bool ok   = (rr != m) && (tsq > 0.f) && (td <= CUTOFF_F) && (w != 0.f);
        if (ok) {
            const float d = pd - td;
            num += d * d;
            den += 1.f;
        }
    }
#pragma unroll
    for (int off = 16; off > 0; off >>= 1) {
        num += __shfl_xor(num, off, 32);
        den += __shfl_xor(den, off, 32);
    }
    __shared__ float sl[8];
    __shared__ float sh[8];
    if (lane == 0) {
        const float has  = (den > 0.f) ? 1.f : 0.f;
        const float loss = (den > 0.f) ? (num / fmaxf(den, 1.f)) : 0.f;
        sl[wave] = loss;
        sh[wave] = has;
    }
    __syncthreads();
    if (threadIdx.x == 0) {
        float L = 0.f, H = 0.f;
        for (int i = 0; i < 8; ++i) { L += sl[i]; H += sh[i]; }
        loss_out[blockIdx.x] = L;
        has_out[blockIdx.x]  = H;
    }
}

// ------------------------ final combine (1 block x 256) ----------------------
__global__ void finalize_kernel(const float* __restrict__ ws, float* __restrict__ out) {
    __shared__ float red[256];
    const int t = threadIdx.x;

    // ligand coord: mean over batches of se/(3*max(cnt,1))
    float v = 0.f;
    if (t < BATCH) v = ws[WS_LSE + t] / (3.f * fmaxf(ws[WS_LCNT + t], 1.f));
    const float ligand_loss = block_sum256(v, red) / (float)BATCH;

    // ligand dist: per batch sum 32 partials (fixed order), then masked mean
    float lossv = 0.f, hasv = 0.f;
    if (t < BATCH) {
        float N = 0.f, D = 0.f;
        for (int i = 0; i < 32; ++i) {
            N += ws[WS_LDN + t * 32 + i];
            D += ws[WS_LDD + t * 32 + i];
        }
        if (D > 0.f) { lossv = N / fmaxf(D, 1.f); hasv = 1.f; }
    }
    const float ld_sum  = block_sum256(lossv, red);
    const float ld_has  = block_sum256(hasv, red);
    const float ligand_dist_loss = ld_sum / fmaxf(ld_has, 1.f);

    // sidechain coord
    float sm = 0.f, sc = 0.f;
    for (int i = t; i < 896; i += 256) {
        sm += ws[WS_SCM + i];
        sc += ws[WS_SCC + i];
    }
    const float sc_mse = block_sum256(sm, red);
    const float sc_cnt = block_sum256(sc, red);
    const float sidechain_loss = sc_mse / fmaxf(sc_cnt, 1.f);

    // sidechain dist
    float dl = 0.f, dh = 0.f;
    for (int i = t; i < 8192; i += 256) {
        dl += ws[WS_SDL + i];
        dh += ws[WS_SDH + i];
    }
    const float sd_sum = block_sum256(dl, red);
    const float sd_has = block_sum256(dh, red);
    const float sidechain_dist_loss = sd_sum / fmaxf(sd_has, 1.f);

    if (t == 0) {
        out[0] = 1.0f * ligand_loss
               + 0.2f * ligand_dist_loss
               + 0.5f * sidechain_loss
               + 0.1f * sidechain_dist_loss;
    }
}

extern "C" void kernel_launch(void* const* d_in, const int* in_sizes, int n_in,
                              void* d_out, int out_size, void* d_ws, size_t ws_size,
                              hipStream_t stream) {
    const float* ligand_pred    = (const float*)d_in[0];
    const float* ligand_tgt     = (const float*)d_in[1];
    const float* sidechain_pred = (const float*)d_in[2];
    const float* sidechain_tgt  = (const float*)d_in[3];
    const unsigned char* ligand_mask = (const unsigned char*)d_in[4];
    const unsigned char* atom_mask   = (const unsigned char*)d_in[5];
    float* ws  = (float*)d_ws;
    float* out = (float*)d_out;

    lig_coord_kernel<<<BATCH, 256, 0, stream>>>(ligand_pred, ligand_tgt, ligand_mask,
                                                ws + WS_LSE, ws + WS_LCNT);
    lig_dist_kernel<<<dim3(32, BATCH), 128, 0, stream>>>(ligand_pred, ligand_tgt,
                                                         ligand_mask,
                                                         ws + WS_LDN, ws + WS_LDD);
    sc_coord_kernel<<<896, 256, 0, stream>>>(sidechain_pred, sidechain_tgt, atom_mask,
                                             ws + WS_SCM, ws + WS_SCC);
    sc_dist_kernel<<<8192, 256, 0, stream>>>(sidechain_pred, sidechain_tgt, atom_mask,
                                             ws + WS_SDL, ws + WS_SDH);
    finalize_kernel<<<1, 256, 0, stream>>>(ws, out);
}